// GCNLayer1_4982162063491
// MI455X (gfx1250) — compile-verified
//
#include <hip/hip_runtime.h>

// ---------------------------------------------------------------------------
// GCN layer for MI455X (gfx1250): Linear(128->128) -> BatchNorm1d(3 nodes,
// batch stats) -> 3x3 adjacency matmul -> ReLU, batched B=262144.
//
// Pass 1: TDM tile DMA + f32 WMMA GEMM, accumulate per-node sum / sumsq.
// Pass 2: tiny finalize -> fold BN + adjacency into 3x3 mix M and bias c.
// Pass 3: TDM + WMMA GEMM again, fused normalize/mix/relu, streamed out.
// ---------------------------------------------------------------------------

typedef __attribute__((ext_vector_type(2))) float        v2f;
typedef __attribute__((ext_vector_type(4))) float        v4f;
typedef __attribute__((ext_vector_type(8))) float        v8f;
typedef __attribute__((ext_vector_type(4))) unsigned int v4u;
typedef __attribute__((ext_vector_type(4))) int          v4i;
typedef __attribute__((ext_vector_type(8))) int          v8i;

#define B_TOT    262144
#define NNODE    3
#define FDIM     128
#define ROWS_TOT (B_TOT * NNODE)          // 786432 GEMM rows
#define HROWS    96                       // rows per block = 32 batches (mult of 3 and 16)
#define HSTR     132                      // padded LDS row stride (floats): TDM pad 4 DW
#define WSTRW    130                      // padded LDS row stride for W (floats): TDM pad 2 DW
#define NWAVES   (HROWS / 16)             // 6 waves, one 16-row tile each
#define TPB      (NWAVES * 32)            // 192 threads
#define NBLK     (ROWS_TOT / HROWS)       // 8192 blocks
#define LDS_H_BYTES (HROWS * HSTR * 4)    // 50688
#define LDS_W_BYTES (FDIM * WSTRW * 4)    // 66560
#define LDS_BYTES   (LDS_H_BYTES + LDS_W_BYTES)   // 117248 -> 2 WGs / 320KB WGP
#define BN_EPS 1e-5f

// ws layout (floats): [0..2]=sum per node, [3..5]=sumsq per node,
//                     [8..16]=M[v][u] row-major, [17..19]=c[v]

__global__ void zero_ws_kernel(float* __restrict__ ws) {
    if (threadIdx.x < 32) ws[threadIdx.x] = 0.0f;
}

// ---------------------------------------------------------------------------
// TDM: DMA a 2D tile (tile_d1 rows x tile_d0 elems, 4B elems, row stride
// stride_d0) from global memory into LDS at lds_off, inserting pad_amount
// DWORDs of LDS padding every 128 DWORDs (pad_interval code 6).
// D# packing per CDNA5 ISA 8.3/8.4: group0 = {count/type/addr}, group1 =
// {data_size, pad, dims, strides}; groups 2/3 zero (<=2D tensor).
// ---------------------------------------------------------------------------
__device__ __forceinline__ void tdm_load_2d(const void* gptr, unsigned lds_off,
                                            unsigned tensor_d0, unsigned tensor_d1,
                                            unsigned stride_d0, unsigned tile_d0,
                                            unsigned tile_d1, unsigned pad_amount_code)
{
    unsigned long long ga = (unsigned long long)(uintptr_t)gptr;
    v4u g0;
    g0.x = 1u;                                               // count=1 (valid user D#)
    g0.y = lds_off;                                          // lds_addr (bytes)
    g0.z = (unsigned)(ga & 0xFFFFFFFFu);                     // global_addr[31:0]
    g0.w = (unsigned)((ga >> 32) & 0x1FFFFFFu) | (2u << 30); // ga[56:32] | type=2
    v8i g1;
    g1[0] = (int)((2u << 16)                 // data_size = 4 bytes
                | (1u << 20)                 // pad_enable
                | (6u << 22)                 // pad_interval: every 128 DWORDs
                | (pad_amount_code << 25));  // pad_amount: (code+1) DWORDs
    g1[1] = (int)(tensor_d0 << 16);                          // td0[15:0]; abar=0
    g1[2] = (int)((tensor_d0 >> 16) | (tensor_d1 << 16));    // td0[31:16], td1[15:0]
    g1[3] = (int)((tensor_d1 >> 16) | (tile_d0 << 16));      // td1[31:16], tile_d0
    g1[4] = (int)tile_d1;                                    // tile_d1; tile_d2=0
    g1[5] = (int)stride_d0;                                  // dim0 stride[31:0]
    g1[6] = 0;                                               // stride hi / dim1 stride
    g1[7] = 0;
    v4i gz = {0, 0, 0, 0};
#if __clang_major__ >= 23
    v8i gz8 = {0, 0, 0, 0, 0, 0, 0, 0};
    __builtin_amdgcn_tensor_load_to_lds(g0, g1, gz, gz, gz8, 0);
#else
    __builtin_amdgcn_tensor_load_to_lds(g0, g1, gz, gz, 0);
#endif
}

// ---------------------------------------------------------------------------
// Shared GEMM body. STATS=true: accumulate per-node sum/sumsq of x=W h + b.
// STATS=false: x -> LDS, then out[b,v,:] = relu(sum_u M[v][u] x[b,u,:] + c[v]).
// ---------------------------------------------------------------------------
template <bool STATS>
__global__ __launch_bounds__(TPB) void gemm_kernel(
    const float* __restrict__ h, const float* __restrict__ W,
    const float* __restrict__ bias, float* __restrict__ ws,
    float* __restrict__ out)
{
    extern __shared__ char smem[];
    float* sh  = (float*)smem;                      // h tile / x tile: HROWS x HSTR
    float* swr = (float*)(smem + LDS_H_BYTES);      // W row-major: FDIM x WSTRW
    __shared__ float redbuf[8];                     // per-node sum/sumsq reduction

    const int  tid     = threadIdx.x;
    const long rowBase = (long)blockIdx.x * HROWS;

    if (STATS) { if (tid < 8) redbuf[tid] = 0.0f; }

    // ---- TDM DMA: stage h tile (96x128, +4 DW pad/row) and W (128x128,
    // +2 DW pad/row) into LDS. One wave issues; TENSORcnt tracks completion.
    if (tid < 32) {
        tdm_load_2d(h + rowBase * FDIM, (unsigned)(uintptr_t)sh,
                    FDIM, HROWS, FDIM, FDIM, HROWS, /*pad=4DW*/ 3);
        tdm_load_2d(W, (unsigned)(uintptr_t)swr,
                    FDIM, FDIM, FDIM, FDIM, FDIM, /*pad=2DW*/ 1);
        __builtin_amdgcn_s_wait_tensorcnt(0);
    }
    __syncthreads();

    const int wave  = tid >> 5;
    const int lane  = tid & 31;
    const int lan16 = lane & 15;
    const int hi    = lane >> 4;                    // lane half selects K 0/1 vs 2/3
    const int rowT  = wave * 16;                    // this wave's 16-row tile

    // bias per (lane, col-tile): C/D layout has N = lane&15 for every VGPR
    float bcol[8];
    #pragma unroll
    for (int c = 0; c < 8; ++c) bcol[c] = bias[c * 16 + lan16];

    v8f acc[8];
    #pragma unroll
    for (int c = 0; c < 8; ++c)
        #pragma unroll
        for (int j = 0; j < 8; ++j) acc[c][j] = 0.0f;

    // Fragment base pointers (single base VGPR + immediate DS offsets).
    const float* ap = sh  + (rowT + lan16) * HSTR + 2 * hi;   // A: row m=lan16
    const float* wl = swr + lan16 * WSTRW + 2 * hi;           // B: col n=lan16

    // ---- K loop: D(16x16) += A(16x4, h rows) x B(4x16, W^T)
    // A frag (ISA 7.12.2): lane m=lan16, VGPR0=K k+2*hi, VGPR1=K k+2*hi+1.
    // B frag: lane n, VGPR0=W[n][k+2*hi]=W^T[k+2*hi][n], VGPR1=next k.
    #pragma unroll 4
    for (int k = 0; k < FDIM; k += 4) {
        v2f av = *(const v2f*)&ap[k];
        #pragma unroll
        for (int c = 0; c < 8; ++c) {
            v2f bv = *(const v2f*)&wl[c * 16 * WSTRW + k];
            acc[c] = __builtin_amdgcn_wmma_f32_16x16x4_f32(
                false, av, false, bv, (short)0, acc[c], false, false);
        }
    }

    // ---- add bias (same column for all 8 accumulator VGPRs of a lane)
    #pragma unroll
    for (int c = 0; c < 8; ++c)
        #pragma unroll
        for (int j = 0; j < 8; ++j) acc[c][j] += bcol[c];

    if (STATS) {
        // element (c,j) sits at row = rowBase + rowT + j + 8*hi (rowBase%3==0)
        float s[3] = {0.f, 0.f, 0.f}, q[3] = {0.f, 0.f, 0.f};
        const int rmod = (rowT + 8 * hi) % 3;
        #pragma unroll
        for (int j = 0; j < 8; ++j) {
            const int n = (rmod + j) % 3;
            float sj = 0.f, qj = 0.f;
            #pragma unroll
            for (int c = 0; c < 8; ++c) {
                float x = acc[c][j];
                sj += x; qj += x * x;
            }
            s[n] += sj; q[n] += qj;
        }
        __syncthreads();   // redbuf initialized, h tile no longer needed
        #pragma unroll
        for (int n = 0; n < 3; ++n) {
            atomicAdd(&redbuf[n], s[n]);
            atomicAdd(&redbuf[3 + n], q[n]);
        }
        __syncthreads();
        if (tid < 6) atomicAdd(&ws[tid], redbuf[tid]);
    } else {
        // ---- write x tile back into this wave's own LDS rows
        #pragma unroll
        for (int c = 0; c < 8; ++c)
            #pragma unroll
            for (int j = 0; j < 8; ++j)
                sh[(rowT + j + 8 * hi) * HSTR + c * 16 + lan16] = acc[c][j];
        __syncthreads();

        // ---- fused BN + 3x3 adjacency mix + ReLU, coalesced b128 stores
        const float m00 = ws[8],  m01 = ws[9],  m02 = ws[10];
        const float m10 = ws[11], m11 = ws[12], m12 = ws[13];
        const float m20 = ws[14], m21 = ws[15], m22 = ws[16];
        const float c0  = ws[17], c1  = ws[18], c2  = ws[19];

        for (int idx = tid; idx < 32 * 32; idx += TPB) {
            const int bb = idx >> 5;                // batch within block
            const int cg = idx & 31;                // group of 4 columns
            v4f x0 = *(const v4f*)&sh[(bb * 3 + 0) * HSTR + cg * 4];
            v4f x1 = *(const v4f*)&sh[(bb * 3 + 1) * HSTR + cg * 4];
            v4f x2 = *(const v4f*)&sh[(bb * 3 + 2) * HSTR + cg * 4];
            v4f o0, o1, o2;
            #pragma unroll
            for (int i = 0; i < 4; ++i) {
                float r0 = m00 * x0[i] + m01 * x1[i] + m02 * x2[i] + c0;
                float r1 = m10 * x0[i] + m11 * x1[i] + m12 * x2[i] + c1;
                float r2 = m20 * x0[i] + m21 * x1[i] + m22 * x2[i] + c2;
                o0[i] = fmaxf(r0, 0.0f);
                o1[i] = fmaxf(r1, 0.0f);
                o2[i] = fmaxf(r2, 0.0f);
            }
            float* op = out + (rowBase + bb * 3) * FDIM + cg * 4;
            *(v4f*)(op)            = o0;
            *(v4f*)(op + FDIM)     = o1;
            *(v4f*)(op + 2 * FDIM) = o2;
        }
    }
}

// ---------------------------------------------------------------------------
// Finalize: fold BN (batch stats) + adjacency counts into M[3][3], c[3].
// ---------------------------------------------------------------------------
__global__ void finalize_kernel(const float* __restrict__ gamma,
                                const float* __restrict__ beta,
                                const int* __restrict__ src,
                                const int* __restrict__ dst,
                                float* __restrict__ ws)
{
    if (threadIdx.x != 0) return;
    const float inv = 1.0f / ((float)B_TOT * (float)FDIM);
    float A[3][3] = {};
    for (int e = 0; e < 8; ++e) A[dst[e]][src[e]] += 1.0f;
    float g[3], d[3];
    for (int u = 0; u < 3; ++u) {
        float mean = ws[u] * inv;
        float var  = ws[3 + u] * inv - mean * mean;
        float rs   = rsqrtf(var + BN_EPS);
        g[u] = gamma[u] * rs;
        d[u] = beta[u] - g[u] * mean;
    }
    for (int v = 0; v < 3; ++v) {
        float cv = 0.0f;
        for (int u = 0; u < 3; ++u) {
            ws[8 + v * 3 + u] = A[v][u] * g[u];
            cv += A[v][u] * d[u];
        }
        ws[17 + v] = cv;
    }
}

extern "C" void kernel_launch(void* const* d_in, const int* in_sizes, int n_in,
                              void* d_out, int out_size, void* d_ws, size_t ws_size,
                              hipStream_t stream) {
    const float* h     = (const float*)d_in[0];
    const float* W     = (const float*)d_in[1];
    const float* bias  = (const float*)d_in[2];
    const float* gamma = (const float*)d_in[3];
    const float* beta  = (const float*)d_in[4];
    const int*   src   = (const int*)d_in[5];
    const int*   dst   = (const int*)d_in[6];
    float* out = (float*)d_out;
    float* ws  = (float*)d_ws;

    zero_ws_kernel<<<dim3(1), dim3(32), 0, stream>>>(ws);
    gemm_kernel<true><<<dim3(NBLK), dim3(TPB), LDS_BYTES, stream>>>(h, W, bias, ws, nullptr);
    finalize_kernel<<<dim3(1), dim3(32), 0, stream>>>(gamma, beta, src, dst, ws);
    gemm_kernel<false><<<dim3(NBLK), dim3(TPB), LDS_BYTES, stream>>>(h, W, bias, ws, out);
}